// TemporalCrossAttention_89111981457439
// MI455X (gfx1250) — compile-verified
//
#include <hip/hip_runtime.h>

typedef __attribute__((ext_vector_type(2))) float v2f;
typedef __attribute__((ext_vector_type(4))) float v4f;
typedef __attribute__((ext_vector_type(8))) float v8f;

#define N_   16
#define T_   32
#define D_   768
#define L_   196
#define BL   16            // l's per block (one wave per l)
#define DC   16            // channels per D-chunk
#define NCH  (D_ / DC)     // 48 chunks (even, needed for 2x-unrolled pipeline)
#define TSTR 20            // t-stride (floats) in Q/K LDS tile (conflict-free frag reads)
#define LSTR (T_ * TSTR + 4)   // 644 per-l stride
#define PSTR 34            // row stride of P (32x32 scores) in LDS
#define PLSTR (T_ * PSTR)  // 1088 per-l P size
#define QOFF 0
#define KOFF (BL * LSTR)             // 10304
#define POFF (2 * BL * LSTR)         // 20608
#define PTQOFF (POFF + BL * PLSTR)   // 38016: pos_q transposed [c][l]
#define PTKOFF (PTQOFF + D_ * BL)    // 50304: pos_k transposed [c][l]
#define SMEM_FLOATS (PTKOFF + D_ * BL)  // 62592 floats = 250368 B (<320KB WGP LDS)

static __device__ __forceinline__ v8f zero8() { v8f z = {0,0,0,0,0,0,0,0}; return z; }

// Load one (16l x 32t x 16c) chunk of q/k into 4 v4f registers (no LDS touch),
// prefetching 2 chunks ahead on the FULL path.
template <bool FULL>
static __device__ __forceinline__ void load_chunk(const float* __restrict__ g,
                                                  int lbase, int c0, int tid, v4f r[4]) {
  const int l4 = (tid & 3) * 4;
  const int group = tid >> 2;
#pragma unroll
  for (int rep = 0; rep < 4; ++rep) {
    int row = group + rep * 128;     // 0..511 -> unique (t, i)
    int t = row & 31;
    int i = row >> 5;
    int c = c0 + i;
    if (FULL) {
      int off = (t * D_ + c) * L_ + lbase + l4;    // 32-bit offset, scalar base
      r[rep] = *(const v4f*)(g + off);
      __builtin_prefetch(g + off + 2 * DC * L_);   // global_prefetch, 2 chunks ahead
    } else {
#pragma unroll
      for (int e = 0; e < 4; ++e) {
        int l = lbase + l4 + e; if (l > L_ - 1) l = L_ - 1;
        r[rep][e] = g[(t * D_ + c) * L_ + l];
      }
    }
  }
}

// Store staged registers + pos (from LDS, transposed [c][l]) into the
// transposed per-l LDS tile dst[l][t*TSTR + i].
static __device__ __forceinline__ void store_chunk(const v4f r[4], const float* smem_pos,
                                                   float* dst, int c0, int tid) {
  const int l4 = (tid & 3) * 4;
  const int group = tid >> 2;
#pragma unroll
  for (int rep = 0; rep < 4; ++rep) {
    int row = group + rep * 128;
    int t = row & 31;
    int i = row >> 5;
    v4f pv = *(const v4f*)&smem_pos[(c0 + i) * BL + l4];  // one b128 LDS read
#pragma unroll
    for (int e = 0; e < 4; ++e)
      dst[(l4 + e) * LSTR + t * TSTR + i] = r[rep][e] + pv[e];
  }
}

static __device__ __forceinline__ void mm_scores(const float* Ql, const float* Kl,
                                                 int m, int hi, v8f acc[2][2]) {
#pragma unroll
  for (int ks = 0; ks < 4; ++ks) {
    int cb = 4 * ks + 2 * hi;
    v2f a0 = *(const v2f*)&Ql[(0  + m) * TSTR + cb];   // A rows t=0..15
    v2f a1 = *(const v2f*)&Ql[(16 + m) * TSTR + cb];   // A rows t=16..31
    v2f b0 = *(const v2f*)&Kl[(0  + m) * TSTR + cb];   // B=K^T cols s=0..15
    v2f b1 = *(const v2f*)&Kl[(16 + m) * TSTR + cb];   // B=K^T cols s=16..31
    acc[0][0] = __builtin_amdgcn_wmma_f32_16x16x4_f32(false, a0, false, b0, (short)0, acc[0][0], false, false);
    acc[0][1] = __builtin_amdgcn_wmma_f32_16x16x4_f32(false, a0, false, b1, (short)0, acc[0][1], false, false);
    acc[1][0] = __builtin_amdgcn_wmma_f32_16x16x4_f32(false, a1, false, b0, (short)0, acc[1][0], false, false);
    acc[1][1] = __builtin_amdgcn_wmma_f32_16x16x4_f32(false, a1, false, b1, (short)0, acc[1][1], false, false);
  }
}

static __device__ __forceinline__ void mm_av_store(const float* Kl, const float* Pl,
                                                   int m, int hi, int n, int lg, int c0,
                                                   float* __restrict__ out) {
  v8f o0 = zero8(), o1 = zero8();
#pragma unroll
  for (int ks = 0; ks < 8; ++ks) {
    int sb = 4 * ks + 2 * hi;
    v2f a0 = *(const v2f*)&Pl[(0  + m) * PSTR + sb];   // A = P rows t=0..15
    v2f a1 = *(const v2f*)&Pl[(16 + m) * PSTR + sb];   // A = P rows t=16..31
    v2f b;                                             // B = K chunk (s x c)
    b.x = Kl[(sb + 0) * TSTR + m];
    b.y = Kl[(sb + 1) * TSTR + m];
    o0 = __builtin_amdgcn_wmma_f32_16x16x4_f32(false, a0, false, b, (short)0, o0, false, false);
    o1 = __builtin_amdgcn_wmma_f32_16x16x4_f32(false, a1, false, b, (short)0, o1, false, false);
  }
  if (lg < L_) {
#pragma unroll
    for (int ti = 0; ti < 2; ++ti) {
      v8f o = ti ? o1 : o0;
#pragma unroll
      for (int r = 0; r < 8; ++r) {
        int t = ti * 16 + hi * 8 + r;
        out[((n * T_ + t) * L_ + lg) * D_ + c0 + m] = o[r];  // 64B coalesced rows
      }
    }
  }
}

template <bool FULL>
static __device__ __forceinline__ void body(const float* __restrict__ qn,
                                            const float* __restrict__ kn,
                                            float* __restrict__ out,
                                            float* smem, int n, int lbase,
                                            int tid, int lane, int w) {
  const int m  = lane & 15;
  const int hi = lane >> 4;
  const float* Ql = smem + QOFF + w * LSTR;
  const float* Kl = smem + KOFF + w * LSTR;
  float* Pl = smem + POFF + w * PLSTR;

  v8f acc[2][2];
#pragma unroll
  for (int ti = 0; ti < 2; ++ti)
#pragma unroll
    for (int si = 0; si < 2; ++si) acc[ti][si] = zero8();

  v4f raq[4], rak[4], rbq[4], rbk[4];

  // ---------------- Pass 1: S = Qp*Kp^T, software-pipelined over 48 chunks ----------------
  load_chunk<FULL>(qn, lbase, 0, tid, raq);
  load_chunk<FULL>(kn, lbase, 0, tid, rak);
  for (int ch = 0; ch < NCH; ch += 2) {
    __syncthreads();                                        // prev compute done (+pos ready)
    store_chunk(raq, smem + PTQOFF, smem + QOFF, ch * DC, tid);
    store_chunk(rak, smem + PTKOFF, smem + KOFF, ch * DC, tid);
    __syncthreads();
    load_chunk<FULL>(qn, lbase, (ch + 1) * DC, tid, rbq);   // overlap with compute below
    load_chunk<FULL>(kn, lbase, (ch + 1) * DC, tid, rbk);
    mm_scores(Ql, Kl, m, hi, acc);
    __syncthreads();
    store_chunk(rbq, smem + PTQOFF, smem + QOFF, (ch + 1) * DC, tid);
    store_chunk(rbk, smem + PTKOFF, smem + KOFF, (ch + 1) * DC, tid);
    __syncthreads();
    if (ch + 2 < NCH) {
      load_chunk<FULL>(qn, lbase, (ch + 2) * DC, tid, raq);
      load_chunk<FULL>(kn, lbase, (ch + 2) * DC, tid, rak);
    }
    mm_scores(Ql, Kl, m, hi, acc);
  }

  // Issue pass-2's first K chunk now; it flies while we do the softmax.
  load_chunk<FULL>(kn, lbase, 0, tid, rak);

  // ---------------- Scores -> LDS, then softmax over s (lane = row t) ----------------
#pragma unroll
  for (int ti = 0; ti < 2; ++ti)
#pragma unroll
    for (int si = 0; si < 2; ++si)
#pragma unroll
      for (int r = 0; r < 8; ++r) {
        int t = ti * 16 + hi * 8 + r;   // C layout: VGPR r -> M = 8*hi + r
        int s = si * 16 + m;            //           N = lane (mod 16)
        Pl[t * PSTR + s] = acc[ti][si][r];
      }
  {
    const float scale = 0.03608439182435161f;  // 1/sqrt(768)
    float* prow = Pl + lane * PSTR;
    float mx = -3.402823466e38f;
#pragma unroll
    for (int s = 0; s < 32; ++s) mx = fmaxf(mx, prow[s]);
    float sum = 0.f;
#pragma unroll
    for (int s = 0; s < 32; ++s) {
      float p = __expf((prow[s] - mx) * scale);
      prow[s] = p;
      sum += p;
    }
    float inv = 1.0f / sum;
#pragma unroll
    for (int s = 0; s < 32; ++s) prow[s] *= inv;
  }

  // ---------------- Pass 2: out = P*Kp, software-pipelined ----------------
  const int lg = lbase + w;
  for (int ch = 0; ch < NCH; ch += 2) {
    __syncthreads();                                        // all waves done with prev K
    store_chunk(rak, smem + PTKOFF, smem + KOFF, ch * DC, tid);
    __syncthreads();
    load_chunk<FULL>(kn, lbase, (ch + 1) * DC, tid, rbk);
    mm_av_store(Kl, Pl, m, hi, n, lg, ch * DC, out);
    __syncthreads();
    store_chunk(rbk, smem + PTKOFF, smem + KOFF, (ch + 1) * DC, tid);
    __syncthreads();
    if (ch + 2 < NCH) load_chunk<FULL>(kn, lbase, (ch + 2) * DC, tid, rak);
    mm_av_store(Kl, Pl, m, hi, n, lg, (ch + 1) * DC, out);
  }
}

__global__ void __launch_bounds__(512, 1)
tca_kernel(const float* __restrict__ q, const float* __restrict__ k,
           const float* __restrict__ pq, const float* __restrict__ pk,
           float* __restrict__ out) {
  extern __shared__ float smem[];
  const int tid  = threadIdx.x;
  const int lane = tid & 31;
  const int w    = tid >> 5;
  const int n    = blockIdx.y;
  const int lbase = blockIdx.x * BL;

  const float* qn = q + (size_t)n * T_ * D_ * L_;
  const float* kn = k + (size_t)n * T_ * D_ * L_;

  // -------- Stage pos embeds once, transposed to [c][l] for b128 reads --------
#pragma unroll
  for (int it = 0; it < 6; ++it) {
    int idx = tid + it * 512;            // 0..3071 quads per matrix
    int ll  = idx / 192;                 // 0..15   (192 c-quads per l)
    int c4  = (idx - ll * 192) * 4;
    int l   = lbase + ll; if (l > L_ - 1) l = L_ - 1;
    v4f vq = *(const v4f*)(pq + l * D_ + c4);
    v4f vk = *(const v4f*)(pk + l * D_ + c4);
#pragma unroll
    for (int e = 0; e < 4; ++e) {
      smem[PTQOFF + (c4 + e) * BL + ll] = vq[e];
      smem[PTKOFF + (c4 + e) * BL + ll] = vk[e];
    }
  }
  // (pos visibility guaranteed by the first __syncthreads inside body's pass-1 loop)

  if (lbase + BL <= L_) body<true >(qn, kn, out, smem, n, lbase, tid, lane, w);
  else                  body<false>(qn, kn, out, smem, n, lbase, tid, lane, w);
}

extern "C" void kernel_launch(void* const* d_in, const int* in_sizes, int n_in,
                              void* d_out, int out_size, void* d_ws, size_t ws_size,
                              hipStream_t stream) {
  const float* q  = (const float*)d_in[0];
  const float* k  = (const float*)d_in[1];
  const float* pq = (const float*)d_in[2];
  const float* pk = (const float*)d_in[3];
  float* out = (float*)d_out;

  const int smem_bytes = SMEM_FLOATS * 4;  // 250368 B
  (void)hipFuncSetAttribute((const void*)tca_kernel,
                            hipFuncAttributeMaxDynamicSharedMemorySize, smem_bytes);

  dim3 grid((L_ + BL - 1) / BL, N_);   // 13 x 16 = 208 blocks, 16 waves each
  dim3 block(512);
  tca_kernel<<<grid, block, smem_bytes, stream>>>(q, k, pq, pk, out);
}